// SegmentAwareLossModule_84713934946708
// MI455X (gfx1250) — compile-verified
//
#include <hip/hip_runtime.h>
#include <hip/hip_bf16.h>

typedef __attribute__((ext_vector_type(2))) float v2f;
typedef __attribute__((ext_vector_type(8))) float v8f;

#define HWIMG 512
#define HW (512 * 512)
#define TS 16
#define HS 18

// charbonnier(x) = (x^2 + 0.001^2)^0.45 ; argument of log2 is >= 1e-6 > 0
__device__ __forceinline__ float charb(float x) {
  return exp2f(0.45f * log2f(x * x + 1e-6f));
}
__device__ __forceinline__ float pow045(float x) {
  return exp2f(0.45f * log2f(x));
}

// zero ws[0..975], plant B-fragment constant quad {1,1,0,0} at ws[976..979]
__global__ void k_init(float* __restrict__ p) {
  int i = blockIdx.x * blockDim.x + threadIdx.x;
  for (; i < 980; i += blockDim.x * gridDim.x)
    p[i] = (i == 976 || i == 977) ? 1.0f : 0.0f;
}

// Pass A: sums[b][s][0..2] = { sum(mask*u), sum(mask*v), sum(mask) }
// via V_WMMA_F32_16X16X4_F32: A = masks (M=16 segments x K=4 pixels),
// B = 4x16 with col0=u, col1=v, col2=1, rest 0 (const cols streamed from a
// stride-0 per-lane pointer so the loop is branch-free and EXEC stays ~0).
__global__ __launch_bounds__(256) void k_seg_sums(
    const float* __restrict__ flow, const float* __restrict__ masks,
    const float* __restrict__ cbuf, float* __restrict__ sums) {
  const int b = blockIdx.y;
  const int lane = threadIdx.x & 31;
  const int wv = threadIdx.x >> 5;
  const int gwave = blockIdx.x * 8 + wv;  // 0..127 per batch
  const int n = lane & 15;                // A: row M (segment); B/D: col N
  const int g = lane >> 4;                // K half-select
  const int PIX = HW / 128;               // 2048 pixels per wave
  const int p0 = gwave * PIX;
  const int pend = p0 + PIX;

  const float* __restrict__ uch = flow + (size_t)b * 2 * HW;
  const float* __restrict__ vch = uch + HW;

  // per-lane A pointer: masks[b][n][p0 + 2g ...]
  const float* aptr = masks + ((size_t)b * 16 + n) * (size_t)HW + p0 + 2 * g;

  // per-lane B pointer + stride (floats): lanes 0/1 walk u/v, lanes >=2 sit
  // on the constant quad (stride 0): cbuf[0..1]={1,1}, cbuf[2..3]={0,0}
  const float* bptr;
  int bs;
  if (n == 0)      { bptr = uch + p0 + 2 * g; bs = 1; }
  else if (n == 1) { bptr = vch + p0 + 2 * g; bs = 1; }
  else if (n == 2) { bptr = cbuf;             bs = 0; }
  else             { bptr = cbuf + 2;         bs = 0; }

  v8f acc0 = {}, acc1 = {};
  for (int p = p0; p < pend; p += 16) {
    const v2f a0 = *(const v2f*)(aptr);
    const v2f a1 = *(const v2f*)(aptr + 4);
    const v2f a2 = *(const v2f*)(aptr + 8);
    const v2f a3 = *(const v2f*)(aptr + 12);
    const v2f b0 = *(const v2f*)(bptr);
    const v2f b1 = *(const v2f*)(bptr + 4 * bs);
    const v2f b2 = *(const v2f*)(bptr + 8 * bs);
    const v2f b3 = *(const v2f*)(bptr + 12 * bs);
    if (p + 1024 < pend) __builtin_prefetch(aptr + 1024, 0, 1);
    acc0 = __builtin_amdgcn_wmma_f32_16x16x4_f32(
        false, a0, false, b0, (short)0, acc0, false, false);
    acc1 = __builtin_amdgcn_wmma_f32_16x16x4_f32(
        false, a1, false, b1, (short)0, acc1, false, false);
    acc0 = __builtin_amdgcn_wmma_f32_16x16x4_f32(
        false, a2, false, b2, (short)0, acc0, false, false);
    acc1 = __builtin_amdgcn_wmma_f32_16x16x4_f32(
        false, a3, false, b3, (short)0, acc1, false, false);
    aptr += 16;
    bptr += 16 * bs;
  }
  const v8f acc = acc0 + acc1;

  // D layout: lane&15 = N (column), VGPR r holds M = r + 8*(lane>>4)
  if (n < 3) {
#pragma unroll
    for (int r = 0; r < 8; ++r) {
      const int m = r + 8 * g;
      atomicAdd(&sums[(b * 16 + m) * 3 + n], acc[r]);
    }
  }
}

// mean_flow, and scale_s[s] = (w_s / max(w_s,eps)) / tw  (0 if tw<=eps)
__global__ void k_finalize(const float* __restrict__ sums,
                           float* __restrict__ mean_u,
                           float* __restrict__ mean_v,
                           float* __restrict__ scale_s) {
  __shared__ float sm[128];
  __shared__ float sw[16];
  __shared__ float stw;
  const int t = threadIdx.x;  // 128 threads: t = b*16 + s
  float msum = fmaxf(sums[t * 3 + 2], 1e-6f);
  mean_u[t] = sums[t * 3 + 0] / msum;
  mean_v[t] = sums[t * 3 + 1] / msum;
  sm[t] = msum;
  __syncthreads();
  if (t < 16) {
    float wsum = 0.0f;
    for (int bb = 0; bb < 8; ++bb) wsum += sm[bb * 16 + t];
    sw[t] = wsum;
  }
  __syncthreads();
  if (t == 0) {
    float tw = 0.0f;
    for (int ss = 0; ss < 16; ++ss) tw += sw[ss];
    stw = tw;
  }
  __syncthreads();
  if (t < 16) {
    const float tw = stw;
    const float wsum = sw[t];
    scale_s[t] = (tw > 1e-6f) ? (wsum / fmaxf(wsum, 1e-6f)) / tw : 0.0f;
  }
}

// Pass B: fused boundary + edge weights + smoothness + segment variance
__global__ __launch_bounds__(256) void k_main(
    const float* __restrict__ flow, const float* __restrict__ image,
    const float* __restrict__ masks, const float* __restrict__ mean_u,
    const float* __restrict__ mean_v, const float* __restrict__ scale_s,
    float* __restrict__ partials) {
  __shared__ float s_tile[HS * HS];
  __shared__ float s_occ[HS * HS];
  __shared__ float s_fu[HS * HS];
  __shared__ float s_fv[HS * HS];
  __shared__ float s_red[8 * 5];

  const int b = blockIdx.z;
  const int w0 = blockIdx.x * TS;
  const int h0 = blockIdx.y * TS;
  const int tid = threadIdx.x;
  const int tx = tid & 15, ty = tid >> 4;
  const int h = h0 + ty, w = w0 + tx;

  const float* __restrict__ uch = flow + (size_t)b * 2 * HW;
  const float* __restrict__ vch = uch + HW;
  const float* __restrict__ mb = masks + (size_t)b * 16 * HW;
  const float* __restrict__ img = image + (size_t)b * 3 * HW;

  for (int cell = tid; cell < HS * HS; cell += 256) s_occ[cell] = 0.0f;

  const float u_c = uch[h * HWIMG + w];
  const float v_c = vch[h * HWIMG + w];

  float sum_per = 0.0f;
  float segc = 0.0f;

  for (int s = 0; s < 16; ++s) {
    __syncthreads();
    const float* __restrict__ src = mb + (size_t)s * HW;
    for (int cell = tid; cell < HS * HS; cell += 256) {
      const int rr = cell / HS, cc = cell % HS;
      const int gh = h0 - 1 + rr, gw = w0 - 1 + cc;
      float val = 0.0f;
      if (gh >= 0 && gh < HWIMG && gw >= 0 && gw < HWIMG)
        val = src[gh * HWIMG + gw];
      s_tile[cell] = val;
      s_occ[cell] += val;
    }
    __syncthreads();

    const float mc = s_tile[(ty + 1) * HS + (tx + 1)];
    float mx = -3.4e38f, mn = 3.4e38f;
#pragma unroll
    for (int dy = -1; dy <= 1; ++dy) {
      const int gh = h + dy;
      if (gh < 0 || gh >= HWIMG) continue;
#pragma unroll
      for (int dx = -1; dx <= 1; ++dx) {
        const int gw = w + dx;
        if (gw < 0 || gw >= HWIMG) continue;
        const float t = s_tile[(ty + 1 + dy) * HS + (tx + 1 + dx)];
        mx = fmaxf(mx, t);
        mn = fminf(mn, t);
      }
    }
    sum_per += fminf(fmaxf(mx - mn, 0.0f), 1.0f);

    const float du = u_c - mean_u[b * 16 + s];
    const float dv = v_c - mean_v[b * 16 + s];
    const float varm = pow045(du * du + dv * dv + 1e-6f);
    segc += varm * mc * scale_s[s];
  }
  __syncthreads();

  // occupancy boundary (3x3 valid-window max - min)
  float omx = -3.4e38f, omn = 3.4e38f;
#pragma unroll
  for (int dy = -1; dy <= 1; ++dy) {
    const int gh = h + dy;
    if (gh < 0 || gh >= HWIMG) continue;
#pragma unroll
    for (int dx = -1; dx <= 1; ++dx) {
      const int gw = w + dx;
      if (gw < 0 || gw >= HWIMG) continue;
      const float t = s_occ[(ty + 1 + dy) * HS + (tx + 1 + dx)];
      omx = fmaxf(omx, t);
      omn = fminf(omn, t);
    }
  }
  const float bocc = fminf(fmaxf(omx - omn, 0.0f), 1.0f);
  const float boundary = fminf(bocc + sum_per, 1.0f);
  const float bweight = 1.0f - 0.9f * boundary;

  // Sobel edge weight; zero halo == zero padding (conv semantics)
  float gmag = 0.0f;
  for (int ch = 0; ch < 3; ++ch) {
    __syncthreads();
    const float* __restrict__ src = img + (size_t)ch * HW;
    for (int cell = tid; cell < HS * HS; cell += 256) {
      const int rr = cell / HS, cc = cell % HS;
      const int gh = h0 - 1 + rr, gw = w0 - 1 + cc;
      float val = 0.0f;
      if (gh >= 0 && gh < HWIMG && gw >= 0 && gw < HWIMG)
        val = src[gh * HWIMG + gw];
      s_tile[cell] = val;
    }
    __syncthreads();
    const int rr = ty + 1, cc = tx + 1;
    const float t00 = s_tile[(rr - 1) * HS + cc - 1];
    const float t01 = s_tile[(rr - 1) * HS + cc];
    const float t02 = s_tile[(rr - 1) * HS + cc + 1];
    const float t10 = s_tile[rr * HS + cc - 1];
    const float t12 = s_tile[rr * HS + cc + 1];
    const float t20 = s_tile[(rr + 1) * HS + cc - 1];
    const float t21 = s_tile[(rr + 1) * HS + cc];
    const float t22 = s_tile[(rr + 1) * HS + cc + 1];
    const float gx = (t00 + 2.0f * t10 + t20) - (t02 + 2.0f * t12 + t22);
    const float gy = (t00 + 2.0f * t01 + t02) - (t20 + 2.0f * t21 + t22);
    gmag += fabsf(gx) + fabsf(gy);
  }
  const float edge_w = expf(-10.0f * (gmag * (1.0f / 3.0f)));
  const float wt = edge_w * bweight;

  // flow tiles for smoothness
  __syncthreads();
  for (int cell = tid; cell < HS * HS; cell += 256) {
    const int rr = cell / HS, cc = cell % HS;
    const int gh = h0 - 1 + rr, gw = w0 - 1 + cc;
    float uu = 0.0f, vv = 0.0f;
    if (gh >= 0 && gh < HWIMG && gw >= 0 && gw < HWIMG) {
      const int o = gh * HWIMG + gw;
      uu = uch[o];
      vv = vch[o];
    }
    s_fu[cell] = uu;
    s_fv[cell] = vv;
  }
  __syncthreads();

  const int rr = ty + 1, cc = tx + 1;
  const float uc2 = s_fu[rr * HS + cc];
  const float vc2 = s_fv[rr * HS + cc];
  float l1x = 0.0f, l1y = 0.0f, l2x = 0.0f, l2y = 0.0f;
  if (w >= 1) {
    const float fxu = uc2 - s_fu[rr * HS + cc - 1];
    const float fxv = vc2 - s_fv[rr * HS + cc - 1];
    l1x = 0.5f * (charb(fxu) + charb(fxv)) * wt;
  }
  if (h >= 1) {
    const float fyu = uc2 - s_fu[(rr - 1) * HS + cc];
    const float fyv = vc2 - s_fv[(rr - 1) * HS + cc];
    l1y = 0.5f * (charb(fyu) + charb(fyv)) * wt;
  }
  if (w >= 1 && w <= HWIMG - 2) {
    const float fxxu = s_fu[rr * HS + cc + 1] - 2.0f * uc2 + s_fu[rr * HS + cc - 1];
    const float fxxv = s_fv[rr * HS + cc + 1] - 2.0f * vc2 + s_fv[rr * HS + cc - 1];
    l2x = charb(fxxu) + charb(fxxv);
  }
  if (h >= 1 && h <= HWIMG - 2) {
    const float fyyu = s_fu[(rr + 1) * HS + cc] - 2.0f * uc2 + s_fu[(rr - 1) * HS + cc];
    const float fyyv = s_fv[(rr + 1) * HS + cc] - 2.0f * vc2 + s_fv[(rr - 1) * HS + cc];
    l2y = charb(fyyu) + charb(fyyv);
  }

  // block reduction (wave32 shuffles), then spread atomics over 64 slots
  float vals[5] = {segc, l1x, l1y, l2x, l2y};
  const int lane = tid & 31, wv = tid >> 5;
#pragma unroll
  for (int i = 0; i < 5; ++i) {
    float x = vals[i];
#pragma unroll
    for (int off = 16; off > 0; off >>= 1) x += __shfl_xor(x, off, 32);
    if (lane == 0) s_red[wv * 5 + i] = x;
  }
  __syncthreads();
  if (tid < 5) {
    float x = 0.0f;
    for (int k = 0; k < 8; ++k) x += s_red[k * 5 + tid];
    const int slot = (blockIdx.x + blockIdx.y * 32 + blockIdx.z * 1024) & 63;
    atomicAdd(&partials[slot * 5 + tid], x);
  }
}

__global__ void k_final(const float* __restrict__ partials,
                        float* __restrict__ out) {
  if (threadIdx.x == 0 && blockIdx.x == 0) {
    float acc[5] = {0.0f, 0.0f, 0.0f, 0.0f, 0.0f};
    for (int i = 0; i < 64; ++i)
      for (int j = 0; j < 5; ++j) acc[j] += partials[i * 5 + j];
    const float N1x = 8.0f * 512.0f * 511.0f;
    const float N1y = 8.0f * 511.0f * 512.0f;
    const float N2x = 8.0f * 2.0f * 512.0f * 510.0f;
    const float N2y = 8.0f * 2.0f * 510.0f * 512.0f;
    const float seg = 0.1f * acc[0];
    const float smooth =
        0.15f * (acc[1] / N1x + acc[2] / N1y +
                 0.5f * (acc[3] / N2x + acc[4] / N2y));
    out[0] = seg + smooth;
  }
}

extern "C" void kernel_launch(void* const* d_in, const int* in_sizes, int n_in,
                              void* d_out, int out_size, void* d_ws,
                              size_t ws_size, hipStream_t stream) {
  (void)in_sizes; (void)n_in; (void)out_size; (void)ws_size;
  const float* flow = (const float*)d_in[0];
  const float* image = (const float*)d_in[1];
  const float* masks = (const float*)d_in[2];
  float* out = (float*)d_out;
  float* ws = (float*)d_ws;

  float* sums = ws;           // 384: [b][s][{u,v,cnt}]
  float* mean_u = ws + 384;   // 128
  float* mean_v = ws + 512;   // 128
  float* scale_s = ws + 640;  // 16
  float* partials = ws + 656; // 320: 64 slots x 5 scalars
  float* cbuf = ws + 976;     // 4: {1,1,0,0} B-fragment constants

  k_init<<<4, 256, 0, stream>>>(ws);
  k_seg_sums<<<dim3(16, 8), 256, 0, stream>>>(flow, masks, cbuf, sums);
  k_finalize<<<1, 128, 0, stream>>>(sums, mean_u, mean_v, scale_s);
  k_main<<<dim3(32, 32, 8), 256, 0, stream>>>(flow, image, masks, mean_u,
                                              mean_v, scale_s, partials);
  k_final<<<1, 64, 0, stream>>>(partials, out);
}